// SuperLoss_21947282882963
// MI455X (gfx1250) — compile-verified
//
#include <hip/hip_runtime.h>
#include <math.h>

// ---------------------------------------------------------------------------
// SuperLoss for MI455X (gfx1250, wave32).
//  K1: tau  — EMA closed-form dot; fp32 weights 0.9^(n-1-k) underflow to 0 for
//      exponents ≳1000, so only a 4096-element tail window contributes (exact
//      fp32 reference semantics). Single block.
//  K2: main — grid-stride float4 stream, Lambert-W Halley (8 iters, single
//      fast divide per iteration), per-block partial sums.
//  K3: final — deterministic reduction of partials, write mean.
// Wave reductions: V_WMMA_F32_16X16X4_F32 (f32 accumulate) + ds_swizzle xor16.
// ---------------------------------------------------------------------------

#define BLOCK 256                       // 8 wave32s per block (all kernels)

typedef __attribute__((ext_vector_type(2))) float v2f;
typedef __attribute__((ext_vector_type(8))) float v8f;

#if __has_builtin(__builtin_amdgcn_wmma_f32_16x16x4_f32)
#define HAVE_WMMA_F32_16X16X4 1
#else
#define HAVE_WMMA_F32_16X16X4 0
#endif

// xor lane swap + add via ds_swizzle (group-of-32 mode: offset=(xor<<10)|0x1f).
// Pattern must be an integer constant expression -> template parameter.
template <int PAT>
__device__ __forceinline__ float swizzle_add(float s) {
  int o = __builtin_amdgcn_ds_swizzle(__float_as_int(s), PAT);
  return s + __int_as_float(o);
}

// Full 32-lane sum, result valid in every lane. Call with EXEC all ones.
__device__ __forceinline__ float wave_reduce_sum(float v) {
#if HAVE_WMMA_F32_16X16X4
  // A (16x4 f32): lane L, VGPR0 = A[L%16][L<16?0:2]; VGPR1 (K=1,3) zeroed.
  // B = ones(4x16)  =>  D[m][n] = s_m + s_{16+m}  (column-independent).
  // Sum of 8 C/D VGPRs: lanes 0-15 -> s_0..7+s_16..23, lanes 16-31 -> rest;
  // single xor-16 swizzle-add completes the 32-lane sum in every lane.
  v2f a; a[0] = v;    a[1] = 0.0f;
  v2f b; b[0] = 1.0f; b[1] = 1.0f;
  v8f c = {};
  c = __builtin_amdgcn_wmma_f32_16x16x4_f32(false, a, false, b,
                                            (short)0, c, false, false);
  float s = ((c[0] + c[1]) + (c[2] + c[3])) + ((c[4] + c[5]) + (c[6] + c[7]));
  return swizzle_add<0x401F>(s);                       // xor 16
#else
  v = swizzle_add<0x401F>(v);                          // xor 16
  v = swizzle_add<0x201F>(v);                          // xor 8
  v = swizzle_add<0x101F>(v);                          // xor 4
  v = swizzle_add<0x081F>(v);                          // xor 2
  v = swizzle_add<0x041F>(v);                          // xor 1
  return v;
#endif
}

// Block-wide sum for a fixed 256-thread block; result valid in EVERY thread.
// Second stage is 8 unrolled LDS broadcast reads — fully uniform, no divergent
// shuffle block, no runtime loop scaffolding.
__device__ __forceinline__ float block_reduce_sum(float v) {
  constexpr int NWAVES = BLOCK / 32;
  __shared__ float red[NWAVES];
  float s = wave_reduce_sum(v);                 // all threads, EXEC full
  if ((threadIdx.x & 31) == 0) red[threadIdx.x >> 5] = s;
  __syncthreads();
  float tot = 0.0f;
  #pragma unroll
  for (int wv = 0; wv < NWAVES; ++wv) tot += red[wv];  // LDS broadcast reads
  return tot;
}

// --------------------------- K1: tau (EMA tail) ----------------------------

__global__ void __launch_bounds__(BLOCK)
superloss_tau_kernel(const float* __restrict__ loss, int n,
                     float* __restrict__ tau_out) {
  const int TAIL = 4096;                 // 0.9^4096 == 0 even as fp32 denormal
  const float LN09 = -0.105360515657826301f;  // ln(0.9)
  int base = (n > TAIL) ? (n - TAIL) : 0;
  float s = 0.0f;
  for (int k = base + (int)threadIdx.x; k < n; k += BLOCK) {
    float m     = (float)(n - 1 - k);
    float decay = __expf(m * LN09);      // fp32 pow semantics (underflow -> 0)
    float wgt   = (k == 0) ? decay : 0.1f * decay;
    s = fmaf(wgt, loss[k], s);
  }
  float tot = block_reduce_sum(s);
  if (threadIdx.x == 0) tau_out[0] = tot;
}

// --------------------------- K2: main elementwise --------------------------

#define HALLEY_ITERS 8   // cubic convergence: fp32-exact by iter <=4; margin x2

__device__ __forceinline__ float superloss_elem(float loss, float tau) {
  const float E     = 2.71828182845904523f;
  const float NEG2E = -0.735758882342884644f;  // -2/e
  float beta  = loss - tau;                    // lam == 1
  float inner = 0.5f * fmaxf(beta, NEG2E);     // >= -1/e
  // init: sqrt branch-point expansion for x<0, log1p elsewhere (as reference)
  float w = (inner < 0.0f)
          ? (-1.0f + sqrtf(fmaxf(2.0f * fmaf(E, inner, 1.0f), 0.0f)))
          : log1pf(inner);
  #pragma unroll
  for (int it = 0; it < HALLEY_ITERS; ++it) {
    float ew  = __expf(w);                     // v_exp_f32 (TRANS)
    float f   = fmaf(w, ew, -inner);
    float wp1 = w + 1.0f;
    // Halley update folded to ONE divide:
    //   dw = f / (ew*wp1 - (w+2)*f/(2*wp1)) = 2*f*wp1 / (2*ew*wp1^2 - (w+2)*f)
    float num = 2.0f * f * wp1;
    float den = fmaf(2.0f * ew * wp1, wp1, -(w + 2.0f) * f);
    float dw  = __fdividef(num, den);          // v_rcp_f32 + mul (TRANS)
    dw = (den == 0.0f) ? f : dw;               // reference's denom==0 -> 1 guard
    w -= dw;
  }
  float ls = -w;                               // log_sigma = -W(inner)
  return fmaf(__expf(ls), beta, ls * ls);      // lam == 1
}

__global__ void __launch_bounds__(BLOCK)
superloss_main_kernel(const float* __restrict__ loss, int n,
                      const float* __restrict__ tau_ptr,
                      float* __restrict__ partial) {
  const float tau = tau_ptr[0];
  const int n4     = n >> 2;
  const int stride = gridDim.x * BLOCK;
  const int gtid   = blockIdx.x * BLOCK + threadIdx.x;
  const float4* __restrict__ l4 = (const float4*)loss;

  float acc = 0.0f;
  for (int i = gtid; i < n4; i += stride) {     // 128-bit streaming loads
    if (i + stride < n4) __builtin_prefetch(&l4[i + stride], 0, 0);
    float4 v = l4[i];
    acc += superloss_elem(v.x, tau);
    acc += superloss_elem(v.y, tau);
    acc += superloss_elem(v.z, tau);
    acc += superloss_elem(v.w, tau);
  }
  // scalar tail (n not multiple of 4); reconverged before block_reduce_sum
  int rem0 = n4 << 2;
  if (gtid < (n - rem0)) acc += superloss_elem(loss[rem0 + gtid], tau);

  float tot = block_reduce_sum(acc);
  if (threadIdx.x == 0) partial[blockIdx.x] = tot;
}

// --------------------------- K3: final reduction ---------------------------

__global__ void __launch_bounds__(BLOCK)
superloss_final_kernel(const float* __restrict__ partial, int nblocks, int n,
                       float* __restrict__ out) {
  float s = 0.0f;
  for (int i = threadIdx.x; i < nblocks; i += BLOCK) s += partial[i];
  float tot = block_reduce_sum(s);
  if (threadIdx.x == 0) out[0] = tot / (float)n;
}

// ---------------------------------------------------------------------------

extern "C" void kernel_launch(void* const* d_in, const int* in_sizes, int n_in,
                              void* d_out, int out_size, void* d_ws, size_t ws_size,
                              hipStream_t stream) {
  const float* loss = (const float*)d_in[0];
  const int n = in_sizes[0];

  float* ws      = (float*)d_ws;
  float* tau     = ws;        // ws[0]
  float* partial = ws + 64;   // cacheline-separated partial sums

  const int NBLOCKS = 4096;   // grid-stride; 16M/4/(4096*256) = 4 iters/thread

  superloss_tau_kernel<<<1, BLOCK, 0, stream>>>(loss, n, tau);
  superloss_main_kernel<<<NBLOCKS, BLOCK, 0, stream>>>(loss, n, tau, partial);
  superloss_final_kernel<<<1, BLOCK, 0, stream>>>(partial, NBLOCKS, n,
                                                  (float*)d_out);
}